// NodeBlock_84069689852537
// MI455X (gfx1250) — compile-verified
//
#include <hip/hip_runtime.h>

#define B_  2
#define N_  50000
#define E_  512000
#define De_ 128
#define Dv_ 128
#define DK_ 256   // De+Dv
#define H_  256
#define Do_ 128
#define SP  260   // padded LDS row stride (floats): 260 % 64 = 4 -> conflict-free

typedef float v2f __attribute__((ext_vector_type(2)));
typedef float v8f __attribute__((ext_vector_type(8)));

// ---------------- kernel 1: zero the agg scratch buffer ----------------
__global__ __launch_bounds__(256) void zero_agg(float* __restrict__ agg, int n4) {
  int i = blockIdx.x * 256 + threadIdx.x;
  if (i < n4) ((float4*)agg)[i] = make_float4(0.f, 0.f, 0.f, 0.f);
}

// ---------------- kernel 2: e2n segment-sum scatter --------------------
// One wave32 per edge: 32 lanes x float4 = 128 features; 4 f32 atomics/lane.
// agg (51.2 MB) is L2-resident, so global_atomic_add_f32 lands in L2.
__global__ __launch_bounds__(256) void scatter_edges(
    const float* __restrict__ edge, const int* __restrict__ recv,
    float* __restrict__ agg) {
  int e = blockIdx.x * 8 + (threadIdx.x >> 5);
  if (e >= B_ * E_) return;
  int lane = threadIdx.x & 31;
  int b = e / E_;
  int r = recv[e];
  float4 v = ((const float4*)(edge + (size_t)e * De_))[lane];
  float* dst = agg + ((size_t)b * N_ + r) * De_ + lane * 4;
  atomicAdd(dst + 0, v.x);
  atomicAdd(dst + 1, v.y);
  atomicAdd(dst + 2, v.z);
  atomicAdd(dst + 3, v.w);
}

// ---------------- kernel 3: fused 2-layer MLP via f32 WMMA -------------
// Block = 256 threads (8 wave32), tile = 64 nodes.
// GEMM1: vt[64x256] @ W1[256x256] (+b1, relu) -> hs[64x256] (LDS)
// GEMM2: hs[64x256] @ W2[256x128] (+b2)       -> out
__global__ __launch_bounds__(256) void node_mlp(
    const float* __restrict__ agg,  const float* __restrict__ node,
    const float* __restrict__ W1,   const float* __restrict__ b1,
    const float* __restrict__ W2,   const float* __restrict__ b2,
    float* __restrict__ out) {
  extern __shared__ float smem[];
  float* vt = smem;            // 64 x SP
  float* hs = smem + 64 * SP;  // 64 x SP

  const int b     = blockIdx.y;
  const int node0 = blockIdx.x * 64;
  const int tid   = threadIdx.x;

  // ---- stage vtin tile: cols [0,128) from agg, [128,256) from node_data
  {
    int row  = tid >> 2;        // 0..63
    int q    = tid & 3;         // quarter of the 256 columns
    int ng   = node0 + row;
    bool ok  = ng < N_;
    const float4* ap = (const float4*)(agg  + ((size_t)b * N_ + ng) * De_);
    const float4* np = (const float4*)(node + ((size_t)b * N_ + ng) * Dv_);
    #pragma unroll
    for (int c4 = 0; c4 < 16; ++c4) {
      int c = q * 64 + c4 * 4;
      float4 v = make_float4(0.f, 0.f, 0.f, 0.f);
      if (ok) v = (c < De_) ? ap[c >> 2] : np[(c - De_) >> 2];
      *(float4*)&vt[row * SP + c] = v;   // row*SP*4 = row*1040, 16B aligned
    }
  }
  __syncthreads();

  const int lane = tid & 31;
  const int wave = tid >> 5;   // 0..7
  const int ln   = lane & 15;  // column / row-in-subtile
  const int kh   = lane >> 4;  // 0: K pair {0,1}; 1: K pair {2,3}

  // ---- GEMM1: each wave owns 2 N-subtiles x 4 M-subtiles
  for (int nt = 0; nt < 2; ++nt) {
    const int n0 = (wave * 2 + nt) * 16;
    v8f acc[4];
    #pragma unroll
    for (int mi = 0; mi < 4; ++mi) { v8f z = {}; acc[mi] = z; }
    const float b1v = b1[n0 + ln];

    for (int ks = 0; ks < 64; ++ks) {
      const int kk = ks * 4 + kh * 2;
      v2f bf;
      bf.x = W1[(kk + 0) * H_ + n0 + ln];
      bf.y = W1[(kk + 1) * H_ + n0 + ln];
      #pragma unroll
      for (int mi = 0; mi < 4; ++mi) {
        v2f af = *(const v2f*)&vt[(mi * 16 + ln) * SP + kk];
        acc[mi] = __builtin_amdgcn_wmma_f32_16x16x4_f32(
            false, af, false, bf, (short)0, acc[mi], false, false);
      }
    }
    #pragma unroll
    for (int mi = 0; mi < 4; ++mi)
      #pragma unroll
      for (int j = 0; j < 8; ++j) {
        float v = fmaxf(acc[mi][j] + b1v, 0.f);          // bias + relu
        hs[(mi * 16 + kh * 8 + j) * SP + n0 + ln] = v;   // C layout: VGPR j -> M=j / j+8
      }
  }
  __syncthreads();

  // ---- GEMM2: each wave owns 1 of 8 N-subtiles (Do=128), 4 M-subtiles
  {
    const int n0 = wave * 16;
    v8f acc[4];
    #pragma unroll
    for (int mi = 0; mi < 4; ++mi) { v8f z = {}; acc[mi] = z; }
    const float b2v = b2[n0 + ln];

    for (int ks = 0; ks < 64; ++ks) {
      const int kk = ks * 4 + kh * 2;
      v2f bf;
      bf.x = W2[(kk + 0) * Do_ + n0 + ln];
      bf.y = W2[(kk + 1) * Do_ + n0 + ln];
      #pragma unroll
      for (int mi = 0; mi < 4; ++mi) {
        v2f af = *(const v2f*)&hs[(mi * 16 + ln) * SP + kk];
        acc[mi] = __builtin_amdgcn_wmma_f32_16x16x4_f32(
            false, af, false, bf, (short)0, acc[mi], false, false);
      }
    }
    #pragma unroll
    for (int mi = 0; mi < 4; ++mi)
      #pragma unroll
      for (int j = 0; j < 8; ++j) {
        int m  = mi * 16 + kh * 8 + j;
        int ng = node0 + m;
        if (ng < N_)
          out[((size_t)b * N_ + ng) * Do_ + n0 + ln] = acc[mi][j] + b2v;
      }
  }
}

// ---------------------------- launcher ---------------------------------
extern "C" void kernel_launch(void* const* d_in, const int* in_sizes, int n_in,
                              void* d_out, int out_size, void* d_ws, size_t ws_size,
                              hipStream_t stream) {
  const float* edge = (const float*)d_in[0];
  const float* node = (const float*)d_in[1];
  const float* W1   = (const float*)d_in[2];
  const float* b1   = (const float*)d_in[3];
  const float* W2   = (const float*)d_in[4];
  const float* b2   = (const float*)d_in[5];
  const int*   recv = (const int*)d_in[6];
  float*       outp = (float*)d_out;
  float*       agg  = (float*)d_ws;   // B*N*De f32 = 51.2 MB (L2-resident)

  (void)in_sizes; (void)n_in; (void)out_size; (void)ws_size;

  // 1) zero agg
  const int n4 = (B_ * N_ * De_) / 4;
  zero_agg<<<(n4 + 255) / 256, 256, 0, stream>>>(agg, n4);

  // 2) scatter edges (8 edges per 256-thread block, one wave per edge)
  scatter_edges<<<(B_ * E_ + 7) / 8, 256, 0, stream>>>(edge, recv, agg);

  // 3) fused MLP with f32 WMMA
  dim3 grid((N_ + 63) / 64, B_);
  const size_t ldsBytes = (size_t)2 * 64 * SP * sizeof(float);  // 133,120 B
  node_mlp<<<grid, 256, ldsBytes, stream>>>(agg, node, W1, b1, W2, b2, outp);
}